// Model_37967510897050
// MI455X (gfx1250) — compile-verified
//
#include <hip/hip_runtime.h>
#include <math.h>

// ---------------- model constants ----------------
#define B_    32
#define S_    512
#define PRED_ 96
#define CIN_  21
#define D_    128
#define NSTATE 16
#define RCONST 0.70710678118654752440f
static const int BAND[4] = {256, 128, 64, 32};

typedef __attribute__((ext_vector_type(2))) float v2f;
typedef __attribute__((ext_vector_type(8))) float v8f;

// Explicit global-address-space pointers so GEMM inner-loop memory ops lower
// to global_load/global_store (LOADcnt/STOREcnt only), never flat_* (which
// also ties DScnt into every wait).
typedef const __attribute__((address_space(1))) float* gcf_t;
typedef const __attribute__((address_space(1))) v2f*   gcv2_t;
typedef __attribute__((address_space(1))) float*       gf_t;

__device__ __forceinline__ gcf_t to_gc(const float* p) {
    return (gcf_t)(unsigned long long)p;
}
__device__ __forceinline__ gf_t to_g(float* p) {
    return (gf_t)(unsigned long long)p;
}

// ---------------- device helpers ----------------
__device__ __forceinline__ float gelu_f(float x) {
    return 0.5f * x * (1.0f + erff(x * 0.70710678118654752f));
}
__device__ __forceinline__ float silu_f(float x) {
    return x / (1.0f + expf(-x));
}
__device__ __forceinline__ float softplus_f(float x) {
    return (x > 20.0f) ? x : log1pf(expf(x));
}

// ---------------- generic strided batched GEMM via V_WMMA_F32_16X16X4_F32 ----
// C[b, m, n] = act( sum_k A[b,m,k] * B[b,k,n] + bias[n] )
//
// One wave owns a 16 x (16*NT) strip of C: the A fragment is loaded once per
// k-step and reused by NT independent WMMA accumulator chains. The k-loop is
// software-pipelined (fragments for step k+1 are prefetched before the wmma
// group of step k issues), so wmma waits cover loads issued a full iteration
// earlier while the next loads are already in flight.
// Out-of-range M rows / N cols are index-CLAMPED (they only affect C rows/cols
// that are never stored) so inner-loop loads are unconditional; only the K%4
// tail uses zero-masked loads. AV/BV select 64-bit vector loads when the
// k-pair is contiguous and 8B alignment is host-proven.
struct GemmDesc {
    const float* A;
    const float* B;
    const float* bias;   // indexed by n, or nullptr
    float*       C;
    long long ab, ar, ac;   // A batch/row/col strides (floats)
    long long bb, br, bc;   // B strides
    long long cb, cr, cc;   // C strides
    int M, N, K;
    int act;                // 0 none, 1 gelu(exact), 2 silu, 3 softplus
};

template <int NT, bool AV, bool BV>
__global__ __launch_bounds__(32) void gemm_wmma_f32(GemmDesc g) {
    const int nspan   = NT << 4;
    const int tiles_n = (g.N + nspan - 1) / nspan;
    const int tm = (blockIdx.x / tiles_n) << 4;
    const int tn = (blockIdx.x % tiles_n) * nspan;
    const long long b = blockIdx.y;

    const int lane = threadIdx.x;        // 0..31, wave32
    const int half = lane >> 4;
    const int lr   = lane & 15;

    const int am  = tm + lr;                       // A row this lane feeds
    const int amc = (am < g.M) ? am : (g.M - 1);   // clamped (safe)

    gcf_t pa = to_gc(g.A) + b * g.ab + (long long)amc * g.ar
                          + (long long)(half << 1) * g.ac;
    gcf_t pbb = to_gc(g.B) + b * g.bb + (long long)(half << 1) * g.br;

    int cn[NT];
    long long coff[NT];
#pragma unroll
    for (int j = 0; j < NT; ++j) {
        cn[j] = tn + (j << 4) + lr;                // B/C column this lane feeds
        const int cnc = (cn[j] < g.N) ? cn[j] : (g.N - 1);
        coff[j] = (long long)cnc * g.bc;
    }

    const long long dA = 4 * g.ac;
    const long long dB = 4 * g.br;

    v8f acc[NT];
#pragma unroll
    for (int j = 0; j < NT; ++j) acc[j] = (v8f){0.f,0.f,0.f,0.f,0.f,0.f,0.f,0.f};

#define LOAD_A(dst)                                                            \
    { if (AV) { (dst) = *(gcv2_t)pa; }                                         \
      else    { (dst).x = pa[0]; (dst).y = pa[g.ac]; } }
#define LOAD_B(dst, j)                                                         \
    { if (BV) { (dst) = *(gcv2_t)(pbb + coff[j]); }                            \
      else    { (dst).x = pbb[coff[j]]; (dst).y = pbb[coff[j] + g.br]; } }

    const int kfull = g.K & ~3;
    if (kfull > 0) {
        v2f afr, bfr[NT];
        LOAD_A(afr);
#pragma unroll
        for (int j = 0; j < NT; ++j) LOAD_B(bfr[j], j);

#pragma unroll 2
        for (int k0 = 0; k0 + 4 < kfull; k0 += 4) {
            v2f afc = afr;
            v2f bfc[NT];
#pragma unroll
            for (int j = 0; j < NT; ++j) bfc[j] = bfr[j];
            pa  += dA;
            pbb += dB;
            LOAD_A(afr);                     // prefetch step k0+4
#pragma unroll
            for (int j = 0; j < NT; ++j) LOAD_B(bfr[j], j);
#pragma unroll
            for (int j = 0; j < NT; ++j)     // consume step k0
                acc[j] = __builtin_amdgcn_wmma_f32_16x16x4_f32(
                    false, afc, false, bfc[j], (short)0, acc[j], false, false);
        }
#pragma unroll
        for (int j = 0; j < NT; ++j)         // last full step
            acc[j] = __builtin_amdgcn_wmma_f32_16x16x4_f32(
                false, afr, false, bfr[j], (short)0, acc[j], false, false);
        pa  += dA;
        pbb += dB;
    }

    if (g.K & 3) {   // K tail (fc0 K=21, dt_proj r=2): zero-masked, runs once
        const int k1 = kfull + (half << 1);
        const bool m0 = (k1     < g.K);
        const bool m1 = (k1 + 1 < g.K);
        v2f af;
        af.x = m0 ? pa[0]    : 0.f;
        af.y = m1 ? pa[g.ac] : 0.f;
#pragma unroll
        for (int j = 0; j < NT; ++j) {
            v2f bf;
            bf.x = m0 ? pbb[coff[j]]        : 0.f;
            bf.y = m1 ? pbb[coff[j] + g.br] : 0.f;
            acc[j] = __builtin_amdgcn_wmma_f32_16x16x4_f32(
                false, af, false, bf, (short)0, acc[j], false, false);
        }
    }
#undef LOAD_A
#undef LOAD_B

    float* Cb = g.C + b * g.cb;
#pragma unroll
    for (int j = 0; j < NT; ++j) {
        if (cn[j] >= g.N) continue;
        float bval = (g.bias != nullptr) ? g.bias[cn[j]] : 0.f;
        gf_t Cp = to_g(Cb) + (long long)cn[j] * g.cc;
#pragma unroll
        for (int v = 0; v < 8; ++v) {
            const int m = tm + (half << 3) + v;
            if (m < g.M) {
                float x = acc[j][v] + bval;
                if (g.act == 1)      x = gelu_f(x);
                else if (g.act == 2) x = silu_f(x);
                else if (g.act == 3) x = softplus_f(x);
                Cp[(long long)m * g.cr] = x;
            }
        }
    }
}

// ---------------- RevIN statistics + normalize ----------------
__global__ void norm_stats_kernel(const float* __restrict__ xe,
                                  float* __restrict__ xc,
                                  float* __restrict__ mean,
                                  float* __restrict__ stdv) {
    const int b = blockIdx.x / CIN_;
    const int c = blockIdx.x % CIN_;
    __shared__ float ssum[256], ssq[256];
    float s = 0.f, q = 0.f;
    for (int t = threadIdx.x; t < S_; t += 256) {
        float v = xe[((long long)b * S_ + t) * CIN_ + c];
        s += v; q += v * v;
    }
    ssum[threadIdx.x] = s; ssq[threadIdx.x] = q;
    __syncthreads();
    for (int o = 128; o > 0; o >>= 1) {
        if (threadIdx.x < o) {
            ssum[threadIdx.x] += ssum[threadIdx.x + o];
            ssq[threadIdx.x]  += ssq[threadIdx.x + o];
        }
        __syncthreads();
    }
    __shared__ float m_s, sd_s;
    if (threadIdx.x == 0) {
        float m   = ssum[0] / (float)S_;
        float var = ssq[0] / (float)S_ - m * m;
        float sd  = sqrtf(var + 1e-5f);
        mean[b * CIN_ + c] = m;
        stdv[b * CIN_ + c] = sd;
        m_s = m; sd_s = sd;
    }
    __syncthreads();
    const float m = m_s, sd = sd_s;
    for (int t = threadIdx.x; t < S_; t += 256) {
        long long i = ((long long)b * S_ + t) * CIN_ + c;
        xc[i] = (xe[i] - m) / sd;
    }
}

// ---------------- Haar DWT / IDWT (one level each) ----------------
__global__ void dwt_level_kernel(const float* __restrict__ x,
                                 float* __restrict__ ap,
                                 float* __restrict__ de,
                                 int L, long long total) {
    long long i = (long long)blockIdx.x * blockDim.x + threadIdx.x;
    if (i >= total) return;
    const int h = L >> 1;
    long long bd = i / h;
    int j = (int)(i % h);
    const float* row = x + bd * (long long)L;
    float e = row[2 * j], o = row[2 * j + 1];
    ap[bd * h + j] = (e + o) * RCONST;
    de[bd * h + j] = (e - o) * RCONST;
}

__global__ void idwt_level_kernel(const float* __restrict__ x,
                                  const float* __restrict__ hi,
                                  float* __restrict__ out,
                                  int Lh, long long total) {
    long long i = (long long)blockIdx.x * blockDim.x + threadIdx.x;
    if (i >= total) return;
    long long bd = i / Lh;
    int j = (int)(i % Lh);
    float a = x[bd * Lh + j], h = hi[bd * Lh + j];
    out[bd * (2LL * Lh) + 2 * j]     = (a + h) * RCONST;
    out[bd * (2LL * Lh) + 2 * j + 1] = (a - h) * RCONST;
}

// ---------------- Mamba depthwise length-2 causal conv + SiLU -------------
__global__ void mamba_conv_kernel(const float* __restrict__ xz,
                                  const float* __restrict__ cw,  // (Li,2)
                                  const float* __restrict__ cb,  // (Li,)
                                  float* __restrict__ xcm,
                                  int Li, long long total) {
    long long i = (long long)blockIdx.x * blockDim.x + threadIdx.x;
    if (i >= total) return;  // total = B*128*Li
    int d = (int)(i % Li);
    long long bt = i / Li;
    int t = (int)(bt % 128);
    long long base = bt * (long long)(2 * Li);
    float cur  = xz[base + d];
    float prev = (t > 0) ? xz[base - 2LL * Li + d] : 0.f;
    float v = cw[2 * d] * prev + cw[2 * d + 1] * cur + cb[d];
    xcm[bt * (long long)Li + d] = silu_f(v);
}

// ---------------- Mamba selective-scan (sequential over t=0..127) ---------
__global__ void mamba_scan_kernel(const float* __restrict__ xcm,   // (B,128,Li)
                                  const float* __restrict__ xdbl,  // (B,128,r+32)
                                  const float* __restrict__ dt,    // (B,128,Li)
                                  const float* __restrict__ A_log, // (Li,16)
                                  const float* __restrict__ Dp,    // (Li,)
                                  const float* __restrict__ xz,    // (B,128,2Li)
                                  float* __restrict__ yb,          // (B,128,Li)
                                  int Li, int r) {
    const int b = blockIdx.x;
    const int d = threadIdx.x;   // blockDim.x == Li (32..256)
    __shared__ float Bs[NSTATE], Cs[NSTATE];

    float A[NSTATE], h[NSTATE];
#pragma unroll
    for (int n = 0; n < NSTATE; ++n) {
        A[n] = -expf(A_log[d * NSTATE + n]);
        h[n] = 0.f;
    }
    const float Dd = Dp[d];
    const int rw = r + 2 * NSTATE;

    for (int t = 0; t < 128; ++t) {
        __syncthreads();
        if (threadIdx.x < 2 * NSTATE) {
            int n = threadIdx.x;
            float v = xdbl[((long long)b * 128 + t) * rw + r + n];
            if (n < NSTATE) Bs[n] = v; else Cs[n - NSTATE] = v;
        }
        __syncthreads();
        long long idx = ((long long)b * 128 + t) * Li + d;
        float dtv = dt[idx];
        float xv  = xcm[idx];
        float y = 0.f;
#pragma unroll
        for (int n = 0; n < NSTATE; ++n) {
            float dA = expf(dtv * A[n]);
            h[n] = dA * h[n] + dtv * Bs[n] * xv;
            y += h[n] * Cs[n];
        }
        float zv = xz[((long long)b * 128 + t) * (2LL * Li) + Li + d];
        yb[idx] = (y + Dd * xv) * silu_f(zv);
    }
}

// ---------------- LayerNorm over D of (res + t), in-place into res --------
__global__ void ln_kernel(float* __restrict__ res,       // (B,D,Li)
                          const float* __restrict__ t,   // (B,128,Li)
                          const float* __restrict__ g,
                          const float* __restrict__ be,
                          int Li) {
    const int b  = blockIdx.x / Li;
    const int li = blockIdx.x % Li;
    const int d  = threadIdx.x;  // 128
    long long idx = ((long long)b * D_ + d) * Li + li;
    float v = res[idx] + t[idx];
    __shared__ float sm[D_], sq[D_];
    sm[d] = v; sq[d] = v * v;
    __syncthreads();
    for (int o = D_ / 2; o > 0; o >>= 1) {
        if (d < o) { sm[d] += sm[d + o]; sq[d] += sq[d + o]; }
        __syncthreads();
    }
    __shared__ float mean_s, rstd_s;
    if (d == 0) {
        float m   = sm[0] / (float)D_;
        float var = sq[0] / (float)D_ - m * m;
        mean_s = m;
        rstd_s = rsqrtf(var + 1e-5f);
    }
    __syncthreads();
    res[idx] = (v - mean_s) * rstd_s * g[d] + be[d];
}

// ---------------- x = gelu(x1 + x2) ----------------
__global__ void add_gelu_kernel(const float* __restrict__ x1,
                                const float* __restrict__ x2,
                                float* __restrict__ x, long long total) {
    long long i = (long long)blockIdx.x * blockDim.x + threadIdx.x;
    if (i >= total) return;
    x[i] = gelu_f(x1[i] + x2[i]);
}

// ---------------- de-normalize output ----------------
__global__ void denorm_kernel(const float* __restrict__ p3,
                              const float* __restrict__ mean,
                              const float* __restrict__ stdv,
                              float* __restrict__ out, long long total) {
    long long i = (long long)blockIdx.x * blockDim.x + threadIdx.x;
    if (i >= total) return;   // total = B*PRED*CIN
    int c = (int)(i % CIN_);
    long long b = (i / CIN_) / PRED_;
    out[i] = p3[i] * stdv[b * CIN_ + c] + mean[b * CIN_ + c];
}

// ---------------- host-side helpers ----------------
static inline void gemm(hipStream_t st,
                        const float* A, const float* Bm, const float* bias, float* C,
                        long long ab, long long ar, long long ac,
                        long long bb, long long br, long long bc,
                        long long cb, long long cr, long long cc,
                        int M, int N, int K, int batch, int act) {
    GemmDesc g{A, Bm, bias, C, ab, ar, ac, bb, br, bc, cb, cr, cc, M, N, K, act};
    // Vector (b64) k-pair loads are legal when k is contiguous and every term
    // of the element offset is even (8-byte alignment provable).
    const bool av = (ac == 1) && ((ar & 1LL) == 0) && ((ab & 1LL) == 0);
    const bool bv = (br == 1) && ((bc & 1LL) == 0) && ((bb & 1LL) == 0);
    const int tiles_m = (M + 15) / 16;
    if (N >= 64) {
        const int tiles = tiles_m * ((N + 63) / 64);
        if (av) {
            if (bv) gemm_wmma_f32<4, true,  true ><<<dim3(tiles, batch), 32, 0, st>>>(g);
            else    gemm_wmma_f32<4, true,  false><<<dim3(tiles, batch), 32, 0, st>>>(g);
        } else {
            if (bv) gemm_wmma_f32<4, false, true ><<<dim3(tiles, batch), 32, 0, st>>>(g);
            else    gemm_wmma_f32<4, false, false><<<dim3(tiles, batch), 32, 0, st>>>(g);
        }
    } else {
        const int tiles = tiles_m * ((N + 15) / 16);
        if (av) {
            if (bv) gemm_wmma_f32<1, true,  true ><<<dim3(tiles, batch), 32, 0, st>>>(g);
            else    gemm_wmma_f32<1, true,  false><<<dim3(tiles, batch), 32, 0, st>>>(g);
        } else {
            if (bv) gemm_wmma_f32<1, false, true ><<<dim3(tiles, batch), 32, 0, st>>>(g);
            else    gemm_wmma_f32<1, false, false><<<dim3(tiles, batch), 32, 0, st>>>(g);
        }
    }
}

struct MambaP {
    const float *A_log, *D, *conv_b, *conv_w, *dt_proj_b, *dt_proj_w,
                *in_proj_w, *out_proj_w, *x_proj_w;
};
struct BlockP {
    const float *conv_b, *conv_w;
    const float *ln_b[4], *ln_g[4];
    MambaP mamba[4];
    const float *sa_b[4], *sa_w[4];
    const float *w1;
    const float *w2[4];
};

extern "C" void kernel_launch(void* const* d_in, const int* in_sizes, int n_in,
                              void* d_out, int out_size, void* d_ws, size_t ws_size,
                              hipStream_t stream) {
    (void)in_sizes; (void)n_in; (void)out_size; (void)ws_size;
    const float* x_enc = (const float*)d_in[0];

    // ---- parameter pytree decode (top-level insertion order; nested dicts
    //      flattened jax-style: sorted keys, lists in order) ----
    BlockP bp[2];
    for (int k = 0; k < 2; ++k) {
        int base = 4 + k * 59;
        bp[k].conv_b = (const float*)d_in[base + 0];
        bp[k].conv_w = (const float*)d_in[base + 1];
        for (int i = 0; i < 4; ++i) bp[k].ln_b[i] = (const float*)d_in[base + 2 + i];
        for (int i = 0; i < 4; ++i) bp[k].ln_g[i] = (const float*)d_in[base + 6 + i];
        for (int j = 0; j < 4; ++j) {
            int mb = base + 10 + 9 * j;
            bp[k].mamba[j].A_log      = (const float*)d_in[mb + 0];
            bp[k].mamba[j].D          = (const float*)d_in[mb + 1];
            bp[k].mamba[j].conv_b     = (const float*)d_in[mb + 2];
            bp[k].mamba[j].conv_w     = (const float*)d_in[mb + 3];
            bp[k].mamba[j].dt_proj_b  = (const float*)d_in[mb + 4];
            bp[k].mamba[j].dt_proj_w  = (const float*)d_in[mb + 5];
            bp[k].mamba[j].in_proj_w  = (const float*)d_in[mb + 6];
            bp[k].mamba[j].out_proj_w = (const float*)d_in[mb + 7];
            bp[k].mamba[j].x_proj_w   = (const float*)d_in[mb + 8];
        }
        for (int i = 0; i < 4; ++i) bp[k].sa_b[i] = (const float*)d_in[base + 46 + i];
        for (int i = 0; i < 4; ++i) bp[k].sa_w[i] = (const float*)d_in[base + 50 + i];
        bp[k].w1 = (const float*)d_in[base + 54];
        for (int i = 0; i < 4; ++i) bp[k].w2[i] = (const float*)d_in[base + 55 + i];
    }
    const float* fc0_b = (const float*)d_in[122];
    const float* fc0_w = (const float*)d_in[123];
    const float* fc1_b = (const float*)d_in[124];
    const float* fc1_w = (const float*)d_in[125];
    const float* fc2_b = (const float*)d_in[126];
    const float* fc2_w = (const float*)d_in[127];
    const float* opj_b = (const float*)d_in[128];
    const float* opj_w = (const float*)d_in[129];

    // ---- workspace carve-out (floats) ----
    float* ws = (float*)d_ws;
    const long long BD = (long long)B_ * D_;
    long long o = 0;
    float* XC   = ws + o; o += (long long)B_ * S_ * CIN_;
    float* MEAN = ws + o; o += B_ * CIN_;
    float* STD  = ws + o; o += B_ * CIN_ + 2;   // keep following buffers 8B-aligned
    float* X    = ws + o; o += BD * S_;
    float* X1   = ws + o; o += BD * S_;
    float* X2   = ws + o; o += BD * S_;
    float* A1   = ws + o; o += BD * 256;
    float* A2   = ws + o; o += BD * 128;
    float* A3   = ws + o; o += BD * 64;
    float* A4   = ws + o; o += BD * 32;      // yl
    float* H[4];
    H[0] = ws + o; o += BD * 256;
    H[1] = ws + o; o += BD * 128;
    H[2] = ws + o; o += BD * 64;
    H[3] = ws + o; o += BD * 32;
    float* OUTFT = ws + o; o += BD * 32;
    float* TB   = ws + o; o += BD * 256;
    float* T2   = ws + o; o += BD * 256;
    float* XZ   = ws + o; o += (long long)B_ * 128 * 512;
    float* XCM  = ws + o; o += (long long)B_ * 128 * 256;
    float* XDBL = ws + o; o += (long long)B_ * 128 * 48;
    float* DTb  = ws + o; o += (long long)B_ * 128 * 256;
    float* YB   = ws + o; o += (long long)B_ * 128 * 256;
    float* MOUT = ws + o; o += (long long)B_ * 128 * 256;
    float* P1   = ws + o; o += (long long)B_ * PRED_ * D_;
    float* P2   = ws + o; o += (long long)B_ * PRED_ * 2 * D_;
    float* P3   = ws + o; o += (long long)B_ * PRED_ * CIN_;

    // ---- 1) RevIN stats + normalize ----
    norm_stats_kernel<<<B_ * CIN_, 256, 0, stream>>>(x_enc, XC, MEAN, STD);

    // ---- 2) fc0 embed, stored transposed as X:(B,D,S) ----
    gemm(stream, XC, fc0_w, fc0_b, X,
         (long long)S_ * CIN_, CIN_, 1,
         0, 1, CIN_,
         (long long)D_ * S_, 1, S_,
         S_, D_, CIN_, B_, 0);

    // ---- 3) two wavelet-mamba blocks ----
    for (int blk = 0; blk < 2; ++blk) {
        const BlockP& p = bp[blk];

        // conv branch: X2[b,e,s] = sum_d conv_w[e,d]*X[b,d,s] + conv_b[e]
        gemm(stream, X, p.conv_w, p.conv_b, X2,
             (long long)D_ * S_, 1, S_,
             0, 1, D_,
             (long long)D_ * S_, 1, S_,
             S_, D_, D_, B_, 0);

        // Haar DWT, 4 levels
        {
            long long t0 = BD * 256;
            dwt_level_kernel<<<(int)((t0 + 255) / 256), 256, 0, stream>>>(X,  A1, H[0], 512, t0);
            long long t1 = BD * 128;
            dwt_level_kernel<<<(int)((t1 + 255) / 256), 256, 0, stream>>>(A1, A2, H[1], 256, t1);
            long long t2 = BD * 64;
            dwt_level_kernel<<<(int)((t2 + 255) / 256), 256, 0, stream>>>(A2, A3, H[2], 128, t2);
            long long t3 = BD * 32;
            dwt_level_kernel<<<(int)((t3 + 255) / 256), 256, 0, stream>>>(A3, A4, H[3], 64,  t3);
        }

        // out_ft = einsum('bix,iox->box', yl, weights1): batched over x (32)
        gemm(stream, A4, p.w1, nullptr, OUTFT,
             1, (long long)D_ * 32, 32,
             1, (long long)D_ * 32, 32,
             1, (long long)D_ * 32, 32,
             B_, D_, D_, 32, 0);

        // per-band processing
        for (int i = 0; i < 4; ++i) {
            const int Li = BAND[i];
            const int r  = Li / 16;
            const MambaP& mp = p.mamba[i];

            // sa: TB[b,d',li] = gelu( sum_d H[b,d,li]*sa_w[d',d] + sa_b[d'] )
            gemm(stream, H[i], p.sa_w[i], p.sa_b[i], TB,
                 (long long)D_ * Li, 1, Li,
                 0, 1, D_,
                 (long long)D_ * Li, 1, Li,
                 Li, D_, D_, B_, 1);

            // weights2 einsum: batched over x (Li)
            gemm(stream, TB, p.w2[i], nullptr, T2,
                 1, (long long)D_ * Li, Li,
                 1, (long long)D_ * Li, Li,
                 1, (long long)D_ * Li, Li,
                 B_, D_, D_, Li, 0);

            // Mamba in_proj: XZ (B,128,2Li)
            gemm(stream, T2, mp.in_proj_w, nullptr, XZ,
                 (long long)128 * Li, Li, 1,
                 0, 1, Li,
                 (long long)128 * 2 * Li, 2 * Li, 1,
                 128, 2 * Li, Li, B_, 0);

            // depthwise causal conv + silu
            long long tot = (long long)B_ * 128 * Li;
            mamba_conv_kernel<<<(int)((tot + 255) / 256), 256, 0, stream>>>(
                XZ, mp.conv_w, mp.conv_b, XCM, Li, tot);

            // x_proj: XDBL (B,128,r+32)
            gemm(stream, XCM, mp.x_proj_w, nullptr, XDBL,
                 (long long)128 * Li, Li, 1,
                 0, 1, Li,
                 (long long)128 * (r + 32), (r + 32), 1,
                 128, r + 32, Li, B_, 0);

            // dt_proj + softplus: DTb (B,128,Li)
            gemm(stream, XDBL, mp.dt_proj_w, mp.dt_proj_b, DTb,
                 (long long)128 * (r + 32), (r + 32), 1,
                 0, 1, r,
                 (long long)128 * Li, Li, 1,
                 128, Li, r, B_, 3);

            // selective scan -> YB
            mamba_scan_kernel<<<B_, Li, 0, stream>>>(
                XCM, XDBL, DTb, mp.A_log, mp.D, XZ, YB, Li, r);

            // out_proj -> MOUT (B,128,Li)
            gemm(stream, YB, mp.out_proj_w, nullptr, MOUT,
                 (long long)128 * Li, Li, 1,
                 0, 1, Li,
                 (long long)128 * Li, Li, 1,
                 128, Li, Li, B_, 0);

            // layernorm over D of (res + t), in-place into H[i]
            ln_kernel<<<B_ * Li, D_, 0, stream>>>(H[i], MOUT, p.ln_g[i], p.ln_b[i], Li);
        }

        // Haar IDWT (reuse approx buffers as stage outputs)
        {
            long long t3 = BD * 32;
            idwt_level_kernel<<<(int)((t3 + 255) / 256), 256, 0, stream>>>(OUTFT, H[3], A3, 32,  t3);
            long long t2 = BD * 64;
            idwt_level_kernel<<<(int)((t2 + 255) / 256), 256, 0, stream>>>(A3,    H[2], A2, 64,  t2);
            long long t1 = BD * 128;
            idwt_level_kernel<<<(int)((t1 + 255) / 256), 256, 0, stream>>>(A2,    H[1], A1, 128, t1);
            long long t0 = BD * 256;
            idwt_level_kernel<<<(int)((t0 + 255) / 256), 256, 0, stream>>>(A1,    H[0], X1, 256, t0);
        }

        // x = gelu(x1 + x2)
        long long totx = BD * S_;
        add_gelu_kernel<<<(int)((totx + 255) / 256), 256, 0, stream>>>(X1, X2, X, totx);
    }

    // ---- 4) head ----
    gemm(stream, X, opj_w, opj_b, P1,
         (long long)D_ * S_, S_, 1,
         0, 1, S_,
         (long long)PRED_ * D_, 1, D_,
         D_, PRED_, S_, B_, 0);

    gemm(stream, P1, fc1_w, fc1_b, P2,
         (long long)PRED_ * D_, D_, 1,
         0, 1, D_,
         (long long)PRED_ * 2 * D_, 2 * D_, 1,
         PRED_, 2 * D_, D_, B_, 1);

    gemm(stream, P2, fc2_w, fc2_b, P3,
         (long long)PRED_ * 2 * D_, 2 * D_, 1,
         0, 1, 2 * D_,
         (long long)PRED_ * CIN_, CIN_, 1,
         PRED_, CIN_, 2 * D_, B_, 0);

    long long toto = (long long)B_ * PRED_ * CIN_;
    denorm_kernel<<<(int)((toto + 255) / 256), 256, 0, stream>>>(
        P3, MEAN, STD, (float*)d_out, toto);
}